// LearnableBSplineActivation_41386304864631
// MI455X (gfx1250) — compile-verified
//
#include <hip/hip_runtime.h>
#include <stdint.h>

// Learnable B-spline activation, degree 3, 5 control points, 9 knots.
// Strategy: the spline is a piecewise cubic in each of the 8 knot spans.
// Kernel 1 (one wave): build the 8x4 polynomial-coefficient table (coeffs in
//   the local coordinate s = x - t_j for conditioning) + scalar params into ws.
// Kernel 2 (streaming, memory-bound): per element -> interval index, one
//   conflict-free ds_load_b128 of the 4 coeffs, Horner (3 FMAs), domain mask.
//   float4 NT loads/stores (268 MB stream > 192 MB L2). Table is pulled into
//   LDS with gfx1250 async global->LDS + s_wait_asynccnt.

#define NCTRL  5
#define DEG    3
#define NKNOT  9
#define NBASE0 (NKNOT - 1)   // 8 degree-0 bases
#define NINT   (NKNOT - 1)   // 8 knot intervals

typedef float v4f __attribute__((ext_vector_type(4)));

__global__ void bspline_build_table(const float* __restrict__ cp,
                                    const float* __restrict__ kn,
                                    float* __restrict__ ws) {
  const int j = (int)threadIdx.x;   // one lane per knot interval
  if (j < NINT) {
    float t[NKNOT];
#pragma unroll
    for (int i = 0; i < NKNOT; ++i) t[i] = kn[i];

    // P[i][d]: coefficient of s^d (s = x - t_j) for basis i at current level,
    // restricted to interval j.
    float P[NBASE0][DEG + 1];
#pragma unroll
    for (int i = 0; i < NBASE0; ++i) {
#pragma unroll
      for (int d = 0; d <= DEG; ++d) P[i][d] = 0.0f;
      P[i][0] = (i == j) ? 1.0f : 0.0f;   // degree-0: indicator of interval i
    }

    // Cox-de Boor on polynomial coefficients (matches reference's d==0 guards).
#pragma unroll
    for (int k = 1; k <= DEG; ++k) {
#pragma unroll
      for (int i = 0; i < NBASE0 - 1; ++i) {
        if (i >= NBASE0 - k) continue;
        float n0 = 0.0f, n1 = 0.0f, n2 = 0.0f, n3 = 0.0f;
        float d1 = t[i + k] - t[i];
        if (d1 != 0.0f) {
          // factor (x - t_i) = s + (t_j - t_i)
          float a = (t[j] - t[i]) / d1, b = 1.0f / d1;
          n0 += a * P[i][0];
          n1 += a * P[i][1] + b * P[i][0];
          n2 += a * P[i][2] + b * P[i][1];
          n3 += a * P[i][3] + b * P[i][2];
        }
        float d2 = t[i + k + 1] - t[i + 1];
        if (d2 != 0.0f) {
          // factor (t_{i+k+1} - x) = (t_{i+k+1} - t_j) - s
          float a = (t[i + k + 1] - t[j]) / d2, b = -1.0f / d2;
          n0 += a * P[i + 1][0];
          n1 += a * P[i + 1][1] + b * P[i + 1][0];
          n2 += a * P[i + 1][2] + b * P[i + 1][1];
          n3 += a * P[i + 1][3] + b * P[i + 1][2];
        }
        P[i][0] = n0; P[i][1] = n1; P[i][2] = n2; P[i][3] = n3;
      }
    }

    // f = sum_i cp[i] * B_i^3 restricted to interval j
    float c0 = 0.0f, c1 = 0.0f, c2 = 0.0f, c3 = 0.0f;
#pragma unroll
    for (int i = 0; i < NCTRL; ++i) {
      float w = cp[i];
      c0 = __builtin_fmaf(w, P[i][0], c0);
      c1 = __builtin_fmaf(w, P[i][1], c1);
      c2 = __builtin_fmaf(w, P[i][2], c2);
      c3 = __builtin_fmaf(w, P[i][3], c3);
    }
    v4f cv = { c0, c1, c2, c3 };
    *(v4f*)(ws + 4 * j) = cv;
  }
  if (j == 0) {
    float h = kn[1] - kn[0];          // uniform for the reference's linspace knots
    ws[32] = kn[0];                   // t0
    ws[33] = 1.0f / h;                // inv_h
    ws[34] = h;                       // h
    ws[35] = kn[NKNOT - 1];           // t8 (upper domain bound, exclusive)
  }
}

__global__ void __launch_bounds__(256)
bspline_eval(const float* __restrict__ xin, float* __restrict__ out,
             const float* __restrict__ ws, int n4, int ntot) {
  __shared__ alignas(16) float tab[40];   // 36 used (8*4 coeffs + 4 scalars)

  const int tid = (int)threadIdx.x;
  if (tid < 32) {
    if (tid < 9) {
      // gfx1250 async global->LDS: 9 lanes x 16B = 144B table fill, no VGPR bounce
      uint32_t lds_off = (uint32_t)(uintptr_t)(&tab[0]) + (uint32_t)tid * 16u;
      const float* src = ws + tid * 4;
      asm volatile("global_load_async_to_lds_b128 %0, %1, off"
                   :: "v"(lds_off), "v"(src) : "memory");
    }
    asm volatile("s_wait_asynccnt 0x0" ::: "memory");
  }
  __syncthreads();

  const float t0 = tab[32], invh = tab[33], h = tab[34], t8 = tab[35];
  const v4f* __restrict__ xv  = (const v4f*)xin;
  v4f* __restrict__       ov  = (v4f*)out;
  const v4f*              tv  = (const v4f*)tab;

  const int stride = (int)(blockDim.x * gridDim.x);
  for (int i = (int)(blockIdx.x * blockDim.x) + tid; i < n4; i += stride) {
    __builtin_prefetch(&xv[i + stride], 0, 1);            // global_prefetch
    v4f v = __builtin_nontemporal_load(&xv[i]);           // global_load_b128, TH=NT
    v4f r;
#pragma unroll
    for (int c = 0; c < 4; ++c) {
      float x  = v[c];
      float u  = (x - t0) * invh;
      float fu = __builtin_floorf(u);
      int   jj = (int)fu;
      jj = jj < 0 ? 0 : (jj > NINT - 1 ? NINT - 1 : jj);  // v_med3-style clamp
      float s  = (u - (float)jj) * h;                     // local coordinate
      v4f  cf  = tv[jj];                                  // conflict-free ds_load_b128
      float y  = __builtin_fmaf(
                   __builtin_fmaf(
                     __builtin_fmaf(cf.w, s, cf.z), s, cf.y), s, cf.x);
      bool ok  = (u >= 0.0f) && (x < t8);                 // outside knot span -> 0
      r[c] = ok ? y : 0.0f;
    }
    __builtin_nontemporal_store(r, &ov[i]);               // global_store_b128, TH=NT
  }

  // scalar tail (ntot not divisible by 4 — not the case here, but be safe)
  for (int i = n4 * 4 + (int)(blockIdx.x * blockDim.x) + tid; i < ntot; i += stride) {
    float x  = xin[i];
    float u  = (x - t0) * invh;
    float fu = __builtin_floorf(u);
    int   jj = (int)fu;
    jj = jj < 0 ? 0 : (jj > NINT - 1 ? NINT - 1 : jj);
    float s  = (u - (float)jj) * h;
    v4f  cf  = tv[jj];
    float y  = __builtin_fmaf(
                 __builtin_fmaf(
                   __builtin_fmaf(cf.w, s, cf.z), s, cf.y), s, cf.x);
    bool ok  = (u >= 0.0f) && (x < t8);
    out[i] = ok ? y : 0.0f;
  }
}

extern "C" void kernel_launch(void* const* d_in, const int* in_sizes, int n_in,
                              void* d_out, int out_size, void* d_ws, size_t ws_size,
                              hipStream_t stream) {
  (void)in_sizes; (void)n_in; (void)ws_size;
  const float* x  = (const float*)d_in[0];
  const float* cp = (const float*)d_in[1];   // control_points [5]
  const float* kn = (const float*)d_in[2];   // knots [9]
  float* out = (float*)d_out;
  float* ws  = (float*)d_ws;                 // needs 144 bytes

  bspline_build_table<<<dim3(1), dim3(32), 0, stream>>>(cp, kn, ws);

  const int ntot = out_size;
  const int n4   = ntot >> 2;
  const int threads = 256;                   // 8 wave32 per block
  long want = ((long)n4 + threads - 1) / threads;
  int blocks = (int)(want < 1 ? 1 : (want > 8192 ? 8192 : want));
  bspline_eval<<<dim3(blocks), dim3(threads), 0, stream>>>(x, out, ws, n4, ntot);
}